// VariationalQuantumClassifier_19370302505089
// MI455X (gfx1250) — compile-verified
//
#include <hip/hip_runtime.h>
#include <hip/hip_bf16.h>

typedef __attribute__((ext_vector_type(2))) float v2f;
typedef __attribute__((ext_vector_type(8))) float v8f;

// ---------------------------------------------------------------------------
// Kernel 1: fold the 8-layer per-qubit 2x2 complex unitary chains (qubits 0,1)
// plus head weights/bias into 5 scalar coefficients K0..K4 in workspace.
//   logit(b) = K0 + K1*cos(x0) + K2*sin(x0) + K3*cos(x1) + K4*sin(x1)
// ---------------------------------------------------------------------------
__global__ void vqc_prep_kernel(const float* __restrict__ params,  // [8,16,3]
                                const float* __restrict__ head_w,  // [1,2]
                                const float* __restrict__ head_b,  // [1]
                                float* __restrict__ K)             // [5] out
{
    if (threadIdx.x != 0 || blockIdx.x != 0) return;

    float Kv[5];
    float k0 = head_b[0];

    for (int q = 0; q < 2; ++q) {
        // Accumulated matrix M (complex), init = identity.
        float Mr[2][2] = {{1.f, 0.f}, {0.f, 1.f}};
        float Mi[2][2] = {{0.f, 0.f}, {0.f, 0.f}};

        for (int l = 0; l < 8; ++l) {
            const float* p = params + (l * 16 + q) * 3;
            float phi = p[0], th = p[1], om = p[2];
            float ct = cosf(0.5f * th), st = sinf(0.5f * th);
            float ap = -0.5f * (phi + om);   // m00 = ct * exp(i*ap)
            float am =  0.5f * (phi - om);   // m01 = -st * exp(i*am)
            float c_ap = cosf(ap), s_ap = sinf(ap);
            float c_am = cosf(am), s_am = sinf(am);

            float Lr[2][2], Li[2][2];
            Lr[0][0] =  ct * c_ap;  Li[0][0] =  ct * s_ap;   // exp(-i(phi+om)/2)*ct
            Lr[0][1] = -st * c_am;  Li[0][1] = -st * s_am;   // -exp(+i(phi-om)/2)*st
            Lr[1][0] =  st * c_am;  Li[1][0] = -st * s_am;   //  exp(-i(phi-om)/2)*st
            Lr[1][1] =  ct * c_ap;  Li[1][1] = -ct * s_ap;   //  exp(+i(phi+om)/2)*ct

            float Nr[2][2], Ni[2][2];
            for (int i = 0; i < 2; ++i)
                for (int j = 0; j < 2; ++j) {
                    Nr[i][j] = Lr[i][0]*Mr[0][j] - Li[i][0]*Mi[0][j]
                             + Lr[i][1]*Mr[1][j] - Li[i][1]*Mi[1][j];
                    Ni[i][j] = Lr[i][0]*Mi[0][j] + Li[i][0]*Mr[0][j]
                             + Lr[i][1]*Mi[1][j] + Li[i][1]*Mr[1][j];
                }
            for (int i = 0; i < 2; ++i)
                for (int j = 0; j < 2; ++j) { Mr[i][j] = Nr[i][j]; Mi[i][j] = Ni[i][j]; }
        }

        // ez(x) = A*cos^2(x/2) + B*sin^2(x/2) - 2*cs*[Im(M00*conj(M01)) - Im(M10*conj(M11))]
        //       = P + Q*cos(x) + R*sin(x)
        float A  = (Mr[0][0]*Mr[0][0] + Mi[0][0]*Mi[0][0])
                 - (Mr[1][0]*Mr[1][0] + Mi[1][0]*Mi[1][0]);
        float Bc = (Mr[0][1]*Mr[0][1] + Mi[0][1]*Mi[0][1])
                 - (Mr[1][1]*Mr[1][1] + Mi[1][1]*Mi[1][1]);
        float im0 = Mi[0][0]*Mr[0][1] - Mr[0][0]*Mi[0][1];  // Im(M00*conj(M01))
        float im1 = Mi[1][0]*Mr[1][1] - Mr[1][0]*Mi[1][1];  // Im(M10*conj(M11))
        float P = 0.5f * (A + Bc);
        float Q = 0.5f * (A - Bc);
        float R = -(im0 - im1);

        float w = head_w[q];
        k0 += w * P;
        Kv[1 + 2*q] = w * Q;
        Kv[2 + 2*q] = w * R;
    }
    Kv[0] = k0;
    for (int i = 0; i < 5; ++i) K[i] = Kv[i];
}

// ---------------------------------------------------------------------------
// Kernel 2: bulk pass. One wave32 per 16-example tile.
// A (16x4 f32, documented layout): lanes 0-15 hold K=0/1 (cos x0, sin x0) for
// rows M=lane; lanes 16-31 hold K=2/3 (cos x1, sin x1) for rows M=lane-16.
// B (4x16): weight vector [K1,K2,K3,K4] replicated across all 16 columns, so
// every column of D equals the logit regardless of B's column mapping.
// C: K0 everywhere. D (documented C/D layout): VGPR r, half h -> row r + 8h.
// ---------------------------------------------------------------------------
__global__ void vqc_main_kernel(const float* __restrict__ X,  // [B,16]
                                const float* __restrict__ K,  // [5]
                                float* __restrict__ out,      // [B]
                                int ntiles)
{
    int tid  = blockIdx.x * blockDim.x + threadIdx.x;
    int wave = tid >> 5;           // wave32
    int lane = tid & 31;
    if (wave >= ntiles) return;    // wave-uniform exit; EXEC stays all-1s below

    int m    = lane & 15;          // row within tile
    int half = lane >> 4;          // 0 -> feature pair of x0, 1 -> x1
    int base = wave * 16;

    float x = X[(size_t)(base + m) * 16 + half];
    float s, c;
    sincosf(x, &s, &c);

    v2f a; a.x = c; a.y = s;       // VGPR0=K{0|2}=cos, VGPR1=K{1|3}=sin

    float K0 = K[0];
    v2f b;
    b.x = half ? K[3] : K[1];      // B VGPR0: K-row 0 (lanes 0-15) / K-row 2 (lanes 16-31)
    b.y = half ? K[4] : K[2];      // B VGPR1: K-row 1 / K-row 3

    v8f cmat;
#pragma unroll
    for (int i = 0; i < 8; ++i) cmat[i] = K0;

    v8f d = __builtin_amdgcn_wmma_f32_16x16x4_f32(
        /*neg_a=*/false, a, /*neg_b=*/false, b,
        /*c_mod=*/(short)0, cmat, /*reuse_a=*/false, /*reuse_b=*/false);

    // Column 0 (any column) of D holds the 16 logits. Lane 0 -> rows 0..7,
    // lane 16 -> rows 8..15; contiguous 32B-aligned float4 stores.
    if (m == 0) {
        float4 lo = make_float4(d[0], d[1], d[2], d[3]);
        float4 hi = make_float4(d[4], d[5], d[6], d[7]);
        float* o = out + base + half * 8;
        *(float4*)(o + 0) = lo;
        *(float4*)(o + 4) = hi;
    }
}

// Scalar cleanup for B % 16 (unused for B = 524288 but kept for robustness).
__global__ void vqc_tail_kernel(const float* __restrict__ X,
                                const float* __restrict__ K,
                                float* __restrict__ out, int start, int B)
{
    int b = start + threadIdx.x;
    if (b >= B) return;
    float x0 = X[(size_t)b * 16 + 0];
    float x1 = X[(size_t)b * 16 + 1];
    float s0, c0, s1, c1;
    sincosf(x0, &s0, &c0);
    sincosf(x1, &s1, &c1);
    out[b] = K[0] + K[1]*c0 + K[2]*s0 + K[3]*c1 + K[4]*s1;
}

extern "C" void kernel_launch(void* const* d_in, const int* in_sizes, int n_in,
                              void* d_out, int out_size, void* d_ws, size_t ws_size,
                              hipStream_t stream)
{
    const float* X      = (const float*)d_in[0];  // [B,16]
    const float* params = (const float*)d_in[1];  // [8,16,3]
    const float* head_w = (const float*)d_in[2];  // [1,2]
    const float* head_b = (const float*)d_in[3];  // [1]
    float* out = (float*)d_out;
    float* K   = (float*)d_ws;                    // 5 floats of scratch

    int B = in_sizes[0] / 16;
    int ntiles = B / 16;
    int rem    = B % 16;

    vqc_prep_kernel<<<1, 32, 0, stream>>>(params, head_w, head_b, K);

    if (ntiles > 0) {
        const int threads = 256;                  // 8 waves per block
        int wavesPerBlock = threads / 32;
        int blocks = (ntiles + wavesPerBlock - 1) / wavesPerBlock;
        vqc_main_kernel<<<blocks, threads, 0, stream>>>(X, K, out, ntiles);
    }
    if (rem > 0) {
        vqc_tail_kernel<<<1, 32, 0, stream>>>(X, K, out, ntiles * 16, B);
    }
}